// Qint_Embedding_44538810860336
// MI455X (gfx1250) — compile-verified
//
#include <hip/hip_runtime.h>

// Qint embedding gather for MI455X (gfx1250, wave32).
//
// out[token, :] = float(W[x[token], :]) * scales[x[token]]
//   x:       [B*S]        int32   (16384 tokens)
//   W:       [VOCAB, DIM] int8    (50257 x 1024  ~ 50 MiB)
//   scales:  [VOCAB]      float32
//   out:     [B*S, DIM]   float32 (64 MiB)
//
// Memory-bound gather: ~84 MB moved -> ~3.6 us floor at 23.3 TB/s.
// Strategy: 64 threads/token, 16 int8 per thread via one b128 load,
// 4x b128 non-temporal stores (stream the write-once output past L2 so
// the 50 MiB table stays L2-resident across graph replays), plus
// global_prefetch_b8 of the next grid-stride iteration's rows.

#define QDIM   1024
#define QVEC   16                  // int8 elements per thread
#define TPT    (QDIM / QVEC)       // 64 threads per token
#define QBLOCK 256                 // 8 wave32s
#define TPB    (QBLOCK / TPT)      // 4 tokens per block-group

typedef __attribute__((ext_vector_type(4))) float v4f;
typedef __attribute__((ext_vector_type(4))) int   v4i;

__global__ __launch_bounds__(QBLOCK) void qint_embedding_gather(
    const int* __restrict__ x,
    const signed char* __restrict__ w,
    const float* __restrict__ scales,
    float* __restrict__ out,
    int n_tokens)
{
    const int tid  = threadIdx.x;
    const int sub  = tid >> 6;          // token slot within group: 0..3
    const int lane = tid & (TPT - 1);   // 0..63, 16 bytes each

    const int groups = (n_tokens + TPB - 1) / TPB;

    for (int g = blockIdx.x; g < groups; g += gridDim.x) {
        const long token = (long)g * TPB + sub;
        if (token >= n_tokens) continue;

        const int row = x[token];

        // Prefetch this thread's 16B slice of next iteration's row
        // (emits global_prefetch_b8; hides the idx->row dependent chain).
        const int gn = g + gridDim.x;
        const long tn = (long)gn * TPB + sub;
        if (tn < n_tokens) {
            const int nrow = x[tn];
            __builtin_prefetch(w + (size_t)nrow * QDIM + (size_t)lane * QVEC, 0, 1);
        }

        const float scale = scales[row];

        // One coalesced 16-byte load of packed int8 (global_load_b128).
        const v4i packed = *reinterpret_cast<const v4i*>(
            w + (size_t)row * QDIM + (size_t)lane * QVEC);

        v4f* dst = reinterpret_cast<v4f*>(
            out + (size_t)token * QDIM + (size_t)lane * QVEC);

        // Unpack 16 int8 -> 16 f32, scale, 4x non-temporal b128 stores.
#pragma unroll
        for (int d = 0; d < 4; ++d) {
            const int word = packed[d];
            v4f f;
            f.x = (float)(signed char)(word);
            f.y = (float)(signed char)(word >> 8);
            f.z = (float)(signed char)(word >> 16);
            f.w = (float)(word >> 24);          // arithmetic shift sign-extends
            f *= scale;
            __builtin_nontemporal_store(f, dst + d);
        }
    }
}

extern "C" void kernel_launch(void* const* d_in, const int* in_sizes, int n_in,
                              void* d_out, int out_size, void* d_ws, size_t ws_size,
                              hipStream_t stream) {
    (void)n_in; (void)out_size; (void)d_ws; (void)ws_size;

    const int*         x      = (const int*)d_in[0];          // [B*S] int32
    const signed char* w      = (const signed char*)d_in[1];  // [VOCAB*DIM] int8
    const float*       scales = (const float*)d_in[2];        // [VOCAB] f32
    float*             out    = (float*)d_out;                // [B*S*DIM] f32

    const int n_tokens = in_sizes[0];                          // 16384
    const int groups   = (n_tokens + TPB - 1) / TPB;           // 4096
    int grid = groups / 2;                                     // depth-2 grid-stride
    if (grid < 1) grid = 1;                                    //  -> prefetch is live

    hipLaunchKernelGGL(qint_embedding_gather, dim3(grid), dim3(QBLOCK), 0, stream,
                       x, w, scales, out, n_tokens);
}